// FeatureEmbedding_48593259987344
// MI455X (gfx1250) — compile-verified
//
#include <hip/hip_runtime.h>

// FeatureEmbedding: out[b,f,:] = x[b,f]*W[f,:] + bias[f,:], with 8 masked
// feature rows per batch row replaced by mask_token. Pure streaming kernel:
// 256 MiB of output stores dominate (roofline ~11us @ 23.3 TB/s). No WMMA —
// there is no contraction; CDNA5 value comes from async LDS staging + NT
// b128 stores instead.

typedef float v4f __attribute__((ext_vector_type(4)));
typedef int   v4i __attribute__((vector_size(4 * sizeof(int))));
// Pointer types the async-to-LDS builtin expects (from hipcc diagnostic):
typedef __attribute__((address_space(1))) v4i* gv4i_p;   // global (AS1)
typedef __attribute__((address_space(3))) v4i* lv4i_p;   // LDS    (AS3)

#ifndef __has_builtin
#define __has_builtin(x) 0
#endif

constexpr int kB  = 2048;   // batch
constexpr int kF  = 64;     // features
constexpr int kD  = 512;    // d_model
constexpr int kNM = 8;      // masks per row
constexpr int kRowsPerBlock = 32;   // batch rows handled by one block
constexpr int kThreads = 256;       // 8 wave32; 128 lanes (4 waves) per row

__global__ __launch_bounds__(kThreads)
void feat_embed_kernel(const float* __restrict__ x,
                       const int*   __restrict__ maskpos,
                       const float* __restrict__ W,
                       const float* __restrict__ bias,
                       const float* __restrict__ mtok,
                       float*       __restrict__ out)
{
    __shared__ v4f lds_w[kD / 4];
    __shared__ v4f lds_b[kD / 4];
    __shared__ v4f lds_m[kD / 4];

    const int t       = threadIdx.x;
    const int f       = blockIdx.x & (kF - 1);          // feature owned by block
    const int bt_base = (blockIdx.x >> 6) * kRowsPerBlock;

    const float* Wrow = W    + (size_t)f * kD;
    const float* Brow = bias + (size_t)f * kD;

    // ---- Stage W[f,:], bias[f,:], mask_token into LDS (6 KB) ----
#if __has_builtin(__builtin_amdgcn_global_load_async_to_lds_b128)
    if (t < kD / 4) {   // 128 lanes x 16B per array
        __builtin_amdgcn_global_load_async_to_lds_b128(
            (gv4i_p)(Wrow + 4 * t), (lv4i_p)&lds_w[t], 0, 0);
        __builtin_amdgcn_global_load_async_to_lds_b128(
            (gv4i_p)(Brow + 4 * t), (lv4i_p)&lds_b[t], 0, 0);
        __builtin_amdgcn_global_load_async_to_lds_b128(
            (gv4i_p)(mtok + 4 * t), (lv4i_p)&lds_m[t], 0, 0);
    }
#if __has_builtin(__builtin_amdgcn_s_wait_asynccnt)
    __builtin_amdgcn_s_wait_asynccnt(0);
#else
    asm volatile("s_wait_asynccnt 0" ::: "memory");
#endif
#else
    for (int i = t; i < kD / 4; i += kThreads) {
        lds_w[i] = *(const v4f*)(Wrow + 4 * i);
        lds_b[i] = *(const v4f*)(Brow + 4 * i);
        lds_m[i] = *(const v4f*)(mtok + 4 * i);
    }
#endif
    __syncthreads();

    // Each thread owns one float4 column of the 512-wide row; the tile data
    // lives in registers for all kRowsPerBlock batch rows.
    const int rh  = t >> 7;        // 0/1: which of the 2 rows this iteration
    const int col = t & 127;       // float4 column index
    const v4f w4 = lds_w[col];
    const v4f b4 = lds_b[col];
    const v4f m4 = lds_m[col];

    for (int r = 0; r < kRowsPerBlock; r += 2) {
        const int bt = bt_base + r + rh;

        const float xs = x[(size_t)bt * kF + f];          // wave-uniform
        const int* mp  = maskpos + (size_t)bt * kNM;
        bool masked = false;
#pragma unroll
        for (int j = 0; j < kNM; ++j) masked |= (mp[j] == f);

        const v4f r4 = masked ? m4 : (xs * w4 + b4);

        float* dst = out + ((size_t)bt * kF + f) * (size_t)kD + col * 4;
        // Output (256 MiB) is write-once, never re-read: bypass L2 residency.
        __builtin_nontemporal_store(r4, (v4f*)dst);
    }
}

extern "C" void kernel_launch(void* const* d_in, const int* in_sizes, int n_in,
                              void* d_out, int out_size, void* d_ws, size_t ws_size,
                              hipStream_t stream) {
    (void)in_sizes; (void)n_in; (void)d_ws; (void)ws_size; (void)out_size;
    const float* x    = (const float*)d_in[0];
    const int*   mp   = (const int*)d_in[1];   // jnp.int64 demotes to int32 (jax default)
    const float* W    = (const float*)d_in[2];
    const float* bias = (const float*)d_in[3];
    const float* mtok = (const float*)d_in[4];
    float* out = (float*)d_out;

    const int grid = kF * (kB / kRowsPerBlock);   // 64 * 64 = 4096 blocks
    feat_embed_kernel<<<grid, kThreads, 0, stream>>>(x, mp, W, bias, mtok, out);
}